// PWD_41188736369331
// MI455X (gfx1250) — compile-verified
//
#include <hip/hip_runtime.h>
#include <stdint.h>

typedef __attribute__((ext_vector_type(2))) float v2f;
typedef __attribute__((ext_vector_type(8))) float v8f;

#define N_PTS 131072
#define M_CTR 2048
#define D_DIM 64
#define KSTEPS (D_DIM / 4)              // 16 chained fp32 WMMAs per tile
#define CHUNK_CTRS 128                  // centroids staged per LDS chunk
#define CHUNK_TILES (CHUNK_CTRS / 16)   // 8 column tiles per chunk
#define NCHUNKS (M_CTR / CHUNK_CTRS)    // 16 chunks
#define LDS_STRIDE 68                   // 64 + 4 pad: bank = 4*l16 + 2*half, conflict-free
#define CHUNK_LDS_FLOATS (CHUNK_CTRS * LDS_STRIDE)   // 8704 floats = 34 KB

// Async-copy one 32KB centroid chunk into bank-padded LDS rows.
// 2 threads per centroid row; each thread moves a 128B half-row as 8 x b128.
__device__ __forceinline__ void copy_chunk_async(const float* __restrict__ c,
                                                 int chunk, float* lds_buf,
                                                 int tid) {
    const int r = tid >> 1;            // centroid row within chunk (0..127)
    const int h = tid & 1;             // which 32-float half of the row
    const char* src =
        (const char*)(c + ((size_t)chunk * CHUNK_CTRS + r) * D_DIM + h * 32);
    uint32_t lds = (uint32_t)(uintptr_t)lds_buf +
                   (uint32_t)((r * LDS_STRIDE + h * 32) * 4);
#pragma unroll
    for (int i = 0; i < 8; ++i) {
        asm volatile("global_load_async_to_lds_b128 %0, %1, off"
                     :: "v"(lds), "v"(src) : "memory");
        src += 16;
        lds += 16;
    }
}

__device__ __forceinline__ void wait_async_lds() {
    asm volatile("s_wait_asynccnt 0x0" ::: "memory");
}

// One wave32 owns a 16-row strip of P. Centroids stream through a
// double-buffered, bank-conflict-free LDS stage via async global->LDS DMA;
// the fp32 WMMA chain computes P.C^T while the next chunk is in flight.
// Min/argmin is fused so the [N,M] distance matrix never exists in memory.
__global__ __launch_bounds__(256) void pwd_argmin_wmma(
    const float* __restrict__ p,     // [N, 64]
    const float* __restrict__ pn,    // [N]
    const float* __restrict__ c,     // [M, 64]
    const float* __restrict__ c_sq,  // [M]
    float* __restrict__ out_d,       // [N]
    int* __restrict__ out_l)         // [N]
{
    __shared__ float smem[2][CHUNK_LDS_FLOATS];

    const int tid  = threadIdx.x;
    const int lane = tid & 31;
    const int wave = tid >> 5;
    const int l16  = lane & 15;      // row (A) / column (B,C/D) within tile
    const int half = lane >> 4;      // K+2 selector (A/B), rows M+8 (C/D)
    const int n0   = (blockIdx.x * 8 + wave) * 16;

    // ---- A fragments (held in VGPRs for the whole kernel) ----
    // ISA 32-bit A 16x4 layout: lanes 0-15 hold K=0/1, lanes 16-31 K=2/3.
    const float* prow = p + (size_t)(n0 + l16) * D_DIM + 2 * half;
    v2f a[KSTEPS];
#pragma unroll
    for (int k = 0; k < KSTEPS; ++k)
        a[k] = *(const v2f*)(prow + 4 * k);

    float best[8];
    int   bi[8];
#pragma unroll
    for (int v = 0; v < 8; ++v) { best[v] = 3.402823e38f; bi[v] = 0; }

    // Prologue: stage chunk 0.
    copy_chunk_async(c, 0, &smem[0][0], tid);
    wait_async_lds();
    __syncthreads();

    for (int q = 0; q < NCHUNKS; ++q) {
        // Kick off DMA for the next chunk into the other buffer. That buffer
        // held chunk q-1, fully consumed before the barrier that ended q-1.
        if (q + 1 < NCHUNKS)
            copy_chunk_async(c, q + 1, &smem[(q + 1) & 1][0], tid);

        const float* bbuf = &smem[q & 1][0];

#pragma unroll
        for (int jj = 0; jj < CHUNK_TILES; ++jj) {
            const int j = q * CHUNK_TILES + jj;

            // B fragments from padded LDS: column m = j*16 + l16.
            // DWORD bank = 4*l16 + 2*half + 4k (mod 64) -> conflict-free b64.
            const float* brow = bbuf + (jj * 16 + l16) * LDS_STRIDE + 2 * half;
            v2f b[KSTEPS];
#pragma unroll
            for (int k = 0; k < KSTEPS; ++k)
                b[k] = *(const v2f*)(brow + 4 * k);   // ds_load_2addr_b64 pairs

            // K=64 accumulation: 16 chained fp32 WMMAs.
            v8f acc = {};
#pragma unroll
            for (int k = 0; k < KSTEPS; ++k)
                acc = __builtin_amdgcn_wmma_f32_16x16x4_f32(
                    false, a[k], false, b[k], (short)0, acc, false, false);

            // Fused epilogue: d = ||c_m||^2 - 2<p,c>; running min/argmin.
            const float cs = c_sq[j * 16 + l16];
            const int   m  = j * 16 + l16;  // strictly increasing per lane
#pragma unroll
            for (int v = 0; v < 8; ++v) {
                float d = fmaf(-2.0f, acc[v], cs);
                if (d < best[v]) { best[v] = d; bi[v] = m; }  // keeps first m on tie
            }
        }

        // Next chunk landed + everyone done with this buffer before reuse.
        wait_async_lds();
        __syncthreads();
    }

    // ---- min/argmin butterfly over the 16 columns in each half-wave ----
    // xor masks 1,2,4,8 never cross the 16-lane half boundary in wave32.
#pragma unroll
    for (int off = 1; off < 16; off <<= 1) {
#pragma unroll
        for (int v = 0; v < 8; ++v) {
            float ov = __shfl_xor(best[v], off, 32);
            int   oi = __shfl_xor(bi[v],  off, 32);
            if (ov < best[v] || (ov == best[v] && oi < bi[v])) {
                best[v] = ov; bi[v] = oi;
            }
        }
    }

    // C/D layout: VGPR v, half h -> row n0 + v + 8*h. Two writer lanes per v.
#pragma unroll
    for (int v = 0; v < 8; ++v) {
        if (l16 == v) {
            const int row = n0 + v + 8 * half;
            out_d[row] = best[v] + pn[row];   // add ||p_n||^2 only at the end
            out_l[row] = bi[v];
        }
    }
}

extern "C" void kernel_launch(void* const* d_in, const int* in_sizes, int n_in,
                              void* d_out, int out_size, void* d_ws, size_t ws_size,
                              hipStream_t stream) {
    const float* p    = (const float*)d_in[0];
    const float* pn   = (const float*)d_in[1];
    const float* c    = (const float*)d_in[2];
    const float* c_sq = (const float*)d_in[3];

    float* out_d = (float*)d_out;             // min_dists [N] f32
    int*   out_l = (int*)d_out + N_PTS;       // min_labels [N] i32 (raw bits)

    dim3 block(256);                          // 8 waves -> 128 rows per block
    dim3 grid(N_PTS / 128);                   // 1024 blocks
    pwd_argmin_wmma<<<grid, block, 0, stream>>>(p, pn, c, c_sq, out_d, out_l);
}